// SupConLossWithPrototype_86105504350694
// MI455X (gfx1250) — compile-verified
//
#include <hip/hip_runtime.h>
#include <hip/hip_bf16.h>

typedef __attribute__((ext_vector_type(2))) float v2f;
typedef __attribute__((ext_vector_type(4))) float v4f;
typedef __attribute__((ext_vector_type(8))) float v8f;

#define KDIM 128
#define INV_TEMP 5.0f
#define ROWS_PER_BLOCK 128
#define JT 64            // columns staged per j-tile
#define LDSTRIDE 132     // 128 + 4 pad -> ds_load_b64 hits all 64 banks once
#define BUFFLOATS (JT * LDSTRIDE)

// ---------------- pass 0: zero the novel counter ----------------
__global__ void supcon_init_kernel(int* count) {
    if (threadIdx.x == 0 && blockIdx.x == 0) *count = 0;
}

// ---------------- pass 1: novel mask + count --------------------
__global__ void supcon_mask_kernel(const int* __restrict__ labels,
                                   const int* __restrict__ proto_labels,
                                   int M, int B,
                                   int* __restrict__ novel,
                                   int* __restrict__ count) {
    int i = blockIdx.x * blockDim.x + threadIdx.x;
    if (i >= M) return;
    int lab = labels[i];
    int isBase = 0;
    for (int b = 0; b < B; ++b) isBase |= (proto_labels[b] == lab);
    int nv = isBase ? 0 : 1;
    novel[i] = nv;
    if (nv) atomicAdd(count, 1);   // integer atomic: deterministic
}

// Issue the async global->LDS copies for one 64-row feature tile.
// ASYNCcnt tracks completion; caller does s_wait_asynccnt + barrier.
__device__ __forceinline__ void async_stage_tile(const float* __restrict__ F, int jb,
                                                 float* dstbase, int tid) {
    #pragma unroll
    for (int i = 0; i < (JT * KDIM / 4) / 256; ++i) {
        int idx = tid + i * 256;           // float4 index within tile
        int r   = idx >> 5;                // KDIM/4 = 32 float4 per row
        int c4  = idx & 31;
        const float* src = F + (size_t)(jb + r) * KDIM + 4 * c4;
        unsigned int lds_addr =
            (unsigned int)(uintptr_t)(dstbase + r * LDSTRIDE + 4 * c4);
        asm volatile("global_load_async_to_lds_b128 %0, %1, off"
                     :: "v"(lds_addr), "v"(src)
                     : "memory");
    }
}

// ---------------- pass 2: fused S-GEMM + reductions + loss ------
__global__ __launch_bounds__(256)
void supcon_main_kernel(const float* __restrict__ F,
                        const int* __restrict__ novel,
                        const int* __restrict__ labels,
                        const float* __restrict__ protos,
                        const int* __restrict__ nov_count,
                        float* __restrict__ partial,
                        int M, int B) {
    extern __shared__ float lds[];             // 2 * BUFFLOATS floats (double buffer)
    __shared__ float rows_E[ROWS_PER_BLOCK];
    __shared__ float rows_T[ROWS_PER_BLOCK];
    __shared__ float cbuf[256];

    const int tid  = threadIdx.x;
    const int wave = tid >> 5;
    const int lane = tid & 31;
    const int half = lane >> 4;                // 0: K lo pair / rows 0-7, 1: K hi pair / rows 8-15
    const int l16  = lane & 15;
    const int rowbase = blockIdx.x * ROWS_PER_BLOCK;
    const int myrow   = rowbase + wave * 16 + l16;   // A-matrix M index for this lane

    // Preload A fragments (this wave's 16 rows, full K) once; reused for all 8192 columns.
    // f32 16x4 A layout: lanes 0-15 hold K = 4s+{0,1}, lanes 16-31 hold K = 4s+{2,3}.
    v2f afrag[KDIM / 4];
    {
        const float* arow = F + (size_t)myrow * KDIM + 2 * half;
        #pragma unroll
        for (int s = 0; s < KDIM / 4; ++s)
            afrag[s] = *(const v2f*)(arow + 4 * s);
    }

    float Eacc[8], Tacc[8];
    #pragma unroll
    for (int v = 0; v < 8; ++v) { Eacc[v] = 0.0f; Tacc[v] = 0.0f; }

    // Prologue: async-stage tile 0 into buffer 0.
    int buf = 0;
    async_stage_tile(F, 0, lds, tid);
    asm volatile("s_wait_asynccnt 0" ::: "memory");
    __syncthreads();

    for (int jb = 0; jb < M; jb += JT) {
        // Kick off the next tile's async copy into the other buffer while we compute.
        if (jb + JT < M)
            async_stage_tile(F, jb + JT, lds + (buf ^ 1) * BUFFLOATS, tid);

        float* cur = lds + buf * BUFFLOATS;
        #pragma unroll
        for (int c = 0; c < JT / 16; ++c) {
            v8f C = {};
            // B (4x16) layout mirrors A with N across lanes; read from LDS.
            const float* brow = &cur[(16 * c + l16) * LDSTRIDE + 2 * half];
            #pragma unroll
            for (int s = 0; s < KDIM / 4; ++s) {
                v2f b = *(const v2f*)(brow + 4 * s);
                C = __builtin_amdgcn_wmma_f32_16x16x4_f32(
                        false, afrag[s], false, b, (short)0, C, false, false);
            }
            // Epilogue: w = novel[j] && (j != i)  (diagonal auto-excluded for base rows)
            const int   gn  = jb + 16 * c + l16;
            const float nfj = (float)novel[gn];
            #pragma unroll
            for (int v = 0; v < 8; ++v) {
                int gm = rowbase + wave * 16 + v + 8 * half;  // C/D layout: row = v + 8*(lane/16)
                float sv = C[v] * INV_TEMP;
                float w  = (gn != gm) ? nfj : 0.0f;
                Eacc[v] += w * __expf(sv);
                Tacc[v] += w * sv;
            }
        }
        // All my async writes to the other buffer done; barrier makes them
        // workgroup-visible and guarantees everyone finished reading `cur`.
        asm volatile("s_wait_asynccnt 0" ::: "memory");
        __syncthreads();
        buf ^= 1;
    }

    // Reduce across the 16 lanes of each half-wave (each half holds distinct rows).
    #pragma unroll
    for (int v = 0; v < 8; ++v) {
        float e = Eacc[v], t = Tacc[v];
        #pragma unroll
        for (int off = 1; off < 16; off <<= 1) {
            e += __shfl_xor(e, off, 16);
            t += __shfl_xor(t, off, 16);
        }
        if (l16 == 0) {
            rows_E[wave * 16 + v + 8 * half] = e;
            rows_T[wave * 16 + v + 8 * half] = t;
        }
    }
    __syncthreads();

    // Restage this block's own 128 rows for the prototype pass (both buffers' space).
    #pragma unroll
    for (int i = 0; i < (ROWS_PER_BLOCK * KDIM / 4) / 256; ++i) {
        int idx = tid + i * 256;
        int r   = idx >> 5;
        int c4  = idx & 31;
        v4f d = *(const v4f*)(F + (size_t)(rowbase + r) * KDIM + 4 * c4);
        *(v4f*)(&lds[r * LDSTRIDE + 4 * c4]) = d;
    }
    __syncthreads();

    float contrib = 0.0f;
    if (tid < ROWS_PER_BLOCK) {
        const int gi  = rowbase + tid;
        const int lab = labels[gi];
        const int isNovel = novel[gi];
        const float* frow = &lds[tid * LDSTRIDE];
        float Psum = 0.0f, expP = 0.0f, pick = 0.0f;
        for (int b = 0; b < B; ++b) {
            const float* pr = protos + (size_t)b * KDIM;
            float dot = 0.0f;
            #pragma unroll
            for (int k4 = 0; k4 < KDIM / 4; ++k4) {
                v4f fv = *(const v4f*)(frow + 4 * k4);
                v4f pv = *(const v4f*)(pr + 4 * k4);
                dot += fv.x * pv.x + fv.y * pv.y + fv.z * pv.z + fv.w * pv.w;
            }
            float p = dot * INV_TEMP;
            Psum += p;
            expP += __expf(p);
            if (b == lab) pick = p;
        }
        const float E = rows_E[tid];
        const float T = rows_T[tid];
        if (isNovel) {
            int nn = *nov_count;
            float cnt  = (float)(nn - 1);
            float scnt = (cnt > 0.0f) ? cnt : 1.0f;
            float den  = E + Psum;                  // exp_sum + raw proto-logit sum (faithful)
            float num  = T - logf(den) * cnt;
            contrib = -(num / scnt);
        } else {
            contrib = -(pick - logf(E + expP));     // exp_nk + exp_pk
        }
    }
    cbuf[tid] = contrib;
    __syncthreads();
    #pragma unroll
    for (int st = 128; st > 0; st >>= 1) {
        if (tid < st) cbuf[tid] += cbuf[tid + st];
        __syncthreads();
    }
    if (tid == 0) partial[blockIdx.x] = cbuf[0];
}

// ---------------- pass 3: deterministic final sum ----------------
__global__ void supcon_final_kernel(const float* __restrict__ partial, int nblocks,
                                    float* __restrict__ out, int M) {
    if (threadIdx.x == 0 && blockIdx.x == 0) {
        float s = 0.0f;
        for (int i = 0; i < nblocks; ++i) s += partial[i];  // fixed order
        out[0] = s / (float)M;
    }
}

extern "C" void kernel_launch(void* const* d_in, const int* in_sizes, int n_in,
                              void* d_out, int out_size, void* d_ws, size_t ws_size,
                              hipStream_t stream) {
    const float* F       = (const float*)d_in[0];   // [M, 128]
    const int*   labels  = (const int*)d_in[1];     // [M]
    const float* protos  = (const float*)d_in[2];   // [B, 128]
    const int*   plabels = (const int*)d_in[3];     // [B]
    const int M = in_sizes[1];
    const int B = in_sizes[3];

    int*   wsi     = (int*)d_ws;
    int*   count   = wsi;                 // [1]
    int*   novel   = wsi + 4;             // [M], 16B aligned
    float* partial = (float*)(wsi + 4 + M);

    float* out = (float*)d_out;

    supcon_init_kernel<<<1, 1, 0, stream>>>(count);
    supcon_mask_kernel<<<(M + 255) / 256, 256, 0, stream>>>(labels, plabels, M, B, novel, count);

    const int nblocks = M / ROWS_PER_BLOCK;
    const size_t shmem = (size_t)2 * BUFFLOATS * sizeof(float);
    supcon_main_kernel<<<nblocks, 256, shmem, stream>>>(F, novel, labels, protos,
                                                        count, partial, M, B);
    supcon_final_kernel<<<1, 1, 0, stream>>>(partial, nblocks, out, M);
}